// MMI_loss_49125835932126
// MI455X (gfx1250) — compile-verified
//
#include <hip/hip_runtime.h>

typedef __attribute__((ext_vector_type(2))) float v2f;
typedef __attribute__((ext_vector_type(8))) float v8f;

#define D_DIM 80
#define KFRAG (D_DIM / 4)   // 20 WMMA K-steps per 16x16 tile
#define EPSN  1e-12f

// ---------------------------------------------------------------------------
// Kernel 1: per-row inverse L2 norms for o (rows [0,N)) and t (rows [N,2N)).
// ---------------------------------------------------------------------------
__global__ __launch_bounds__(256) void mmi_norms_kernel(const float* __restrict__ o,
                                                        const float* __restrict__ t,
                                                        float* __restrict__ inv_o,
                                                        float* __restrict__ inv_t,
                                                        int N) {
    const int lane = threadIdx.x & 31;
    const int wave = threadIdx.x >> 5;
    const int row  = blockIdx.x * 8 + wave;      // 0 .. 2N-1
    if (row >= 2 * N) return;

    const bool   is_o = (row < N);
    const int    r    = is_o ? row : row - N;
    const float* p    = (is_o ? o : t) + (size_t)r * D_DIM;

    float v0 = p[lane];
    float v1 = p[lane + 32];
    float s  = v0 * v0 + v1 * v1;
    if (lane < 16) { float v2 = p[lane + 64]; s += v2 * v2; }

    s += __shfl_xor(s, 16, 32);
    s += __shfl_xor(s, 8, 32);
    s += __shfl_xor(s, 4, 32);
    s += __shfl_xor(s, 2, 32);
    s += __shfl_xor(s, 1, 32);

    if (lane == 0) {
        float invn = 1.0f / fmaxf(sqrtf(s), EPSN);
        (is_o ? inv_o : inv_t)[r] = invn;
    }
}

// ---------------------------------------------------------------------------
// Kernel 2: fused sim-tile GEMM (fp32 WMMA 16x16x4) + row-wise sum(exp) +
// diagonal extraction. One block (8 waves) owns a 16-row tile; wave w walks
// column tiles jt = w, w+8, ... with double-buffered B fragments.
// - Normalization applied AFTER accumulation (mathematically identical):
//   load -> wmma path stays VALU-free.
// - 4 independent accumulator chains break the WMMA D->C dependency so
//   consecutive v_wmma_f32_16x16x4_f32 issue without delay_alu bubbles.
// ---------------------------------------------------------------------------
__global__ __launch_bounds__(256) void mmi_sim_loss_kernel(const float* __restrict__ o,
                                                           const float* __restrict__ t,
                                                           const float* __restrict__ inv_o,
                                                           const float* __restrict__ inv_t,
                                                           float* __restrict__ block_partials,
                                                           int ntiles) {
    __shared__ float lds_rowsum[8 * 16];
    __shared__ float lds_pos[16];
    __shared__ float lds_loss[16];

    const int lane  = threadIdx.x & 31;
    const int wave  = threadIdx.x >> 5;
    const int i0    = blockIdx.x * 16;
    const int mrow  = lane & 15;          // tile-local row (A) / col-row (B)
    const int khalf = lane >> 4;          // 0: K={0,1}, 1: K={2,3}

    const v2f* ob = reinterpret_cast<const v2f*>(o);
    const v2f* tb = reinterpret_cast<const v2f*>(t);

    // ---- A fragments (unscaled), resident for the whole block
    const v2f* ap = ob + (size_t)(i0 + mrow) * (D_DIM / 2) + khalf;
    v2f a[KFRAG];
    #pragma unroll
    for (int kk = 0; kk < KFRAG; ++kk)
        a[kk] = ap[kk * 2];

    // ---- per-lane row scales: D-frag element r belongs to row i0 + 8*khalf + r
    float ao[8];
    const float* aop = inv_o + i0 + 8 * khalf;
    #pragma unroll
    for (int r = 0; r < 8; ++r)
        ao[r] = aop[r];

    v8f sumexp = {};

    auto loadB = [&](v2f (&buf)[KFRAG], int jtile) {
        const v2f* bp = tb + (size_t)(jtile * 16 + mrow) * (D_DIM / 2) + khalf;
        #pragma unroll
        for (int kk = 0; kk < KFRAG; ++kk)
            buf[kk] = bp[kk * 2];
    };

    auto step = [&](int jt, v2f (&bcur)[KFRAG], v2f (&bnext)[KFRAG], int jnext) {
        const float bt = inv_t[jt * 16 + mrow];   // this lane's column scale
        loadB(bnext, jnext);                      // prefetch next tile first

        // 4 independent accumulator chains (5 WMMAs each), fixed combine order
        v8f acc0 = {}, acc1 = {}, acc2 = {}, acc3 = {};
        #pragma unroll
        for (int q = 0; q < KFRAG / 4; ++q) {
            acc0 = __builtin_amdgcn_wmma_f32_16x16x4_f32(
                false, a[4 * q + 0], false, bcur[4 * q + 0], (short)0, acc0, false, false);
            acc1 = __builtin_amdgcn_wmma_f32_16x16x4_f32(
                false, a[4 * q + 1], false, bcur[4 * q + 1], (short)0, acc1, false, false);
            acc2 = __builtin_amdgcn_wmma_f32_16x16x4_f32(
                false, a[4 * q + 2], false, bcur[4 * q + 2], (short)0, acc2, false, false);
            acc3 = __builtin_amdgcn_wmma_f32_16x16x4_f32(
                false, a[4 * q + 3], false, bcur[4 * q + 3], (short)0, acc3, false, false);
        }
        v8f acc = (acc0 + acc1) + (acc2 + acc3);

        // normalize after accumulation: sim = acc * inv_o[row] * inv_t[col]
        v8f scl;
        #pragma unroll
        for (int r = 0; r < 8; ++r)
            scl[r] = acc[r] * (ao[r] * bt);

        // Diagonal tile: lanes 0-7 hold rows 0-7 at r=lane; lanes 24-31 hold
        // rows 8-15 at r=lane-24.
        if (jt == (int)blockIdx.x) {
            int dr = -1, drow = 0;
            if (lane < 8)        { dr = lane;      drow = lane; }
            else if (lane >= 24) { dr = lane - 24; drow = lane - 16; }
            if (dr >= 0) {
                float dval = 0.0f;
                #pragma unroll
                for (int r = 0; r < 8; ++r)
                    if (r == dr) dval = scl[r];
                lds_pos[drow] = dval;
            }
        }

        #pragma unroll
        for (int r = 0; r < 8; ++r)
            sumexp[r] += __expf(scl[r]);
    };

    // ---- double-buffered walk over this wave's 64 column tiles
    v2f b0[KFRAG], b1[KFRAG];
    loadB(b0, wave);
    for (int jt = wave; jt < ntiles; jt += 16) {
        step(jt,     b0, b1, jt + 8);
        const int jnn = (jt + 16 < ntiles) ? jt + 16 : jt;   // clamp (redundant load)
        step(jt + 8, b1, b0, jnn);
    }

    // ---- deterministic reduction: butterfly within each 16-lane half
    #pragma unroll
    for (int r = 0; r < 8; ++r) {
        float s = sumexp[r];
        s += __shfl_xor(s, 1, 32);
        s += __shfl_xor(s, 2, 32);
        s += __shfl_xor(s, 4, 32);
        s += __shfl_xor(s, 8, 32);
        sumexp[r] = s;
    }
    if ((lane & 15) == 0) {
        const int halfbase = khalf * 8;
        #pragma unroll
        for (int r = 0; r < 8; ++r)
            lds_rowsum[wave * 16 + halfbase + r] = sumexp[r];
    }
    __syncthreads();

    if (threadIdx.x < 16) {
        float tot = 0.0f;
        #pragma unroll
        for (int w = 0; w < 8; ++w)
            tot += lds_rowsum[w * 16 + threadIdx.x];
        lds_loss[threadIdx.x] = logf(tot) - lds_pos[threadIdx.x];
    }
    __syncthreads();

    if (threadIdx.x == 0) {
        float s = 0.0f;
        #pragma unroll
        for (int m = 0; m < 16; ++m)
            s += lds_loss[m];
        block_partials[blockIdx.x] = s;
    }
}

// ---------------------------------------------------------------------------
// Kernel 3: fixed-order final reduction -> mean loss (MMI_WEIGHT = 1).
// ---------------------------------------------------------------------------
__global__ void mmi_finalize_kernel(const float* __restrict__ partials,
                                    float* __restrict__ out, int ntiles, int N) {
    if (blockIdx.x == 0 && threadIdx.x == 0) {
        float s = 0.0f;
        for (int i = 0; i < ntiles; ++i)
            s += partials[i];
        out[0] = s / (float)N;
    }
}

extern "C" void kernel_launch(void* const* d_in, const int* in_sizes, int n_in,
                              void* d_out, int out_size, void* d_ws, size_t ws_size,
                              hipStream_t stream) {
    const float* o = (const float*)d_in[0];   // mel_outputs [16,512,80] fp32
    const float* t = (const float*)d_in[1];   // mel_targets [16,512,80] fp32

    const int N      = in_sizes[0] / D_DIM;   // 8192 rows
    const int ntiles = N / 16;                // 512 16-row tiles

    float* ws       = (float*)d_ws;
    float* inv_o    = ws;                     // [N]
    float* inv_t    = ws + N;                 // [N]
    float* partials = ws + 2 * N;             // [ntiles]

    mmi_norms_kernel<<<(2 * N + 7) / 8, 256, 0, stream>>>(o, t, inv_o, inv_t, N);
    mmi_sim_loss_kernel<<<ntiles, 256, 0, stream>>>(o, t, inv_o, inv_t, partials, ntiles);
    mmi_finalize_kernel<<<1, 1, 0, stream>>>(partials, (float*)d_out, ntiles, N);
}